// VN_Attention_35880156790882
// MI455X (gfx1250) — compile-verified
//
#include <hip/hip_runtime.h>

// Problem constants (from the reference)
#define BDIM 8
#define CDIM 64
#define UDIM 3
#define NDIM 512
#define MDIM (UDIM * NDIM)                     // 1536 = flattened (u,n)
#define TENSOR_ELEMS (BDIM * CDIM * UDIM * NDIM) // 786432 floats = 3 MB

typedef __attribute__((ext_vector_type(2))) float v2f;
typedef __attribute__((ext_vector_type(8))) float v8f;

// D = A(16x4 f32) * B(4x16 f32) + C(16x16 f32), full fp32 matrix op on CDNA5.
__device__ __forceinline__ v8f wmma_f32_16x16x4(v2f a, v2f b, v8f c) {
  return __builtin_amdgcn_wmma_f32_16x16x4_f32(
      /*neg_a=*/false, a, /*neg_b=*/false, b,
      /*c_mod=*/(short)0, c, /*reuse_a=*/false, /*reuse_b=*/false);
}

// ---------------------------------------------------------------------------
// Kernel 1: q/k/v = W @ x  over the channel dim.
// Y[b][d][m] = sum_c W[d][c] * X[b][c][m],  m in [0, 1536)
// grid = (MDIM/64, B, 3 weights), block = 256 threads (8 waves).
// Each wave owns two 16x16 D-subtiles; K=64 loop of v_wmma_f32_16x16x4_f32.
// ---------------------------------------------------------------------------
#define WS 68   // LDS row stride (floats) for W tile
#define XS 72   // LDS row stride (floats) for X tile (breaks half-wave bank alias)

__global__ __launch_bounds__(256) void vn_qkv_proj_kernel(
    const float* __restrict__ x,
    const float* __restrict__ Wq, const float* __restrict__ Wk,
    const float* __restrict__ Wv, float* __restrict__ ws) {
  __shared__ float sW[64 * WS];
  __shared__ float sX[64 * XS];

  const int b  = blockIdx.y;
  const int m0 = blockIdx.x * 64;
  const float* __restrict__ W =
      (blockIdx.z == 0) ? Wq : ((blockIdx.z == 1) ? Wk : Wv);
  float* __restrict__ out = ws + (size_t)blockIdx.z * TENSOR_ELEMS;

  // Stage W (64x64) and the X tile (64 c-rows x 64 m-cols) into LDS.
  const int tid = threadIdx.x;
  for (int i = tid; i < 64 * 16; i += 256) {     // 1024 float4 chunks
    const int r  = i >> 4;
    const int c4 = (i & 15) << 2;
    const float4 w4 = *(const float4*)(W + r * 64 + c4);
    sW[r * WS + c4 + 0] = w4.x; sW[r * WS + c4 + 1] = w4.y;
    sW[r * WS + c4 + 2] = w4.z; sW[r * WS + c4 + 3] = w4.w;
    const float4 x4 =
        *(const float4*)(x + (size_t)b * CDIM * MDIM + (size_t)r * MDIM + m0 + c4);
    sX[r * XS + c4 + 0] = x4.x; sX[r * XS + c4 + 1] = x4.y;
    sX[r * XS + c4 + 2] = x4.z; sX[r * XS + c4 + 3] = x4.w;
  }
  __syncthreads();

  const int lane = tid & 31;
  const int wave = tid >> 5;
  const int l16  = lane & 15;
  const int h    = lane >> 4;            // half-wave select (K split per ISA layout)
  const int msub = (wave & 3) * 16;      // 16-col subtile within the 64-col tile
  const int dA   = (wave >> 2) * 32;     // this wave's two 16-row subtiles
  const int dB   = dA + 16;

  v8f acc0 = {};
  v8f acc1 = {};
#pragma unroll
  for (int k0 = 0; k0 < 64; k0 += 4) {
    const int kk = k0 + 2 * h;           // lanes 0-15: K=k0,k0+1 ; lanes 16-31: K=k0+2,k0+3
    v2f bx, a0, a1;
    bx.x = sX[kk * XS + msub + l16];
    bx.y = sX[(kk + 1) * XS + msub + l16];
    a0.x = sW[(dA + l16) * WS + kk];
    a0.y = sW[(dA + l16) * WS + kk + 1];
    a1.x = sW[(dB + l16) * WS + kk];
    a1.y = sW[(dB + l16) * WS + kk + 1];
    acc0 = wmma_f32_16x16x4(a0, bx, acc0);
    acc1 = wmma_f32_16x16x4(a1, bx, acc1);
  }

  // D layout: VGPR g -> row g (lanes 0-15) / row g+8 (lanes 16-31), col = l16.
  const int m = m0 + msub + l16;
  float* __restrict__ outb = out + (size_t)b * CDIM * MDIM;
#pragma unroll
  for (int g = 0; g < 8; ++g) {
    const int row = g + 8 * h;
    outb[(size_t)(dA + row) * MDIM + m] = acc0[g];
    outb[(size_t)(dB + row) * MDIM + m] = acc1[g];
  }
}

// ---------------------------------------------------------------------------
// Kernel 2: fused attention per (b,c) pair. grid = B*C blocks, 256 threads.
// S = (scale*Q)^T K via v_wmma_f32_16x16x4_f32 (u padded 3->4),
// streaming (flash) softmax with per-row stats kept per 16-lane half,
// P*V^T done as lane-partial FMAs (output width is only 3), residual fused.
// ---------------------------------------------------------------------------
#define KS 520  // LDS row stride for q/k/v panels (2*520 mod 64 == 16 -> no alias)

__global__ __launch_bounds__(256) void vn_attn_kernel(
    const float* __restrict__ x, const float* __restrict__ qg,
    const float* __restrict__ kg, const float* __restrict__ vg,
    float* __restrict__ out) {
  __shared__ float sq[UDIM * KS];
  __shared__ float sk[UDIM * KS];
  __shared__ float sv[UDIM * KS];

  const int bc = blockIdx.x;                       // b*C + c
  const size_t base = (size_t)bc * (UDIM * NDIM);
  const int tid = threadIdx.x;

  // Stage q,k,v [3][512] panels into LDS (float4 chunks).
  for (int i = tid; i < UDIM * (NDIM / 4); i += 256) {
    const int u  = i / (NDIM / 4);
    const int n4 = (i % (NDIM / 4)) * 4;
    float4 t;
    t = *(const float4*)(qg + base + u * NDIM + n4);
    sq[u * KS + n4 + 0] = t.x; sq[u * KS + n4 + 1] = t.y;
    sq[u * KS + n4 + 2] = t.z; sq[u * KS + n4 + 3] = t.w;
    t = *(const float4*)(kg + base + u * NDIM + n4);
    sk[u * KS + n4 + 0] = t.x; sk[u * KS + n4 + 1] = t.y;
    sk[u * KS + n4 + 2] = t.z; sk[u * KS + n4 + 3] = t.w;
    t = *(const float4*)(vg + base + u * NDIM + n4);
    sv[u * KS + n4 + 0] = t.x; sv[u * KS + n4 + 1] = t.y;
    sv[u * KS + n4 + 2] = t.z; sv[u * KS + n4 + 3] = t.w;
  }
  __syncthreads();

  const int lane = tid & 31;
  const int wave = tid >> 5;
  const int l16  = lane & 15;
  const int h    = lane >> 4;
  const float scale = 0.125f;  // C^-0.5 = 64^-0.5

  for (int it = wave; it < NDIM / 16; it += 8) {   // wave-uniform i-tiles
    const int i0 = it * 16;

    // A fragment of (scale*Q)^T [16 i x 4 u] (u=3 padded with 0):
    v2f aq;
    aq.x = scale * (h ? sq[2 * KS + i0 + l16] : sq[0 * KS + i0 + l16]);
    aq.y = h ? 0.0f : scale * sq[1 * KS + i0 + l16];

    float mrow[8], lrow[8], acc0[8], acc1[8], acc2[8];
#pragma unroll
    for (int g = 0; g < 8; ++g) {
      mrow[g] = -1e30f; lrow[g] = 0.0f;
      acc0[g] = 0.0f; acc1[g] = 0.0f; acc2[g] = 0.0f;
    }

    for (int jt = 0; jt < NDIM / 16; ++jt) {
      const int j0 = jt * 16;
      // B fragment of K [4 u x 16 j]:
      v2f bk;
      bk.x = h ? sk[2 * KS + j0 + l16] : sk[0 * KS + j0 + l16];
      bk.y = h ? 0.0f : sk[1 * KS + j0 + l16];
      v8f zero = {};
      v8f S = wmma_f32_16x16x4(aq, bk, zero);      // S[i0+g(+8h)][j0+l16]

      const float v0 = sv[0 * KS + j0 + l16];      // v[u][j_lane]
      const float v1 = sv[1 * KS + j0 + l16];
      const float v2 = sv[2 * KS + j0 + l16];

#pragma unroll
      for (int g = 0; g < 8; ++g) {
        float s  = S[g];
        float sm = s;                               // row max across 16-lane half
        sm = fmaxf(sm, __shfl_xor(sm, 8, 16));
        sm = fmaxf(sm, __shfl_xor(sm, 4, 16));
        sm = fmaxf(sm, __shfl_xor(sm, 2, 16));
        sm = fmaxf(sm, __shfl_xor(sm, 1, 16));
        const float newm = fmaxf(mrow[g], sm);
        const float corr = __expf(mrow[g] - newm);  // uniform across the half
        const float p    = __expf(s - newm);        // per-lane probability
        mrow[g] = newm;
        lrow[g] = lrow[g] * corr + p;               // lane-partial row sum
        acc0[g] = acc0[g] * corr + p * v0;          // lane-partial P*V^T
        acc1[g] = acc1[g] * corr + p * v1;
        acc2[g] = acc2[g] * corr + p * v2;
      }
    }

    // Final cross-lane reductions + normalized write with fused residual.
#pragma unroll
    for (int g = 0; g < 8; ++g) {
      float ls = lrow[g], a0 = acc0[g], a1 = acc1[g], a2 = acc2[g];
#pragma unroll
      for (int off = 8; off > 0; off >>= 1) {
        ls += __shfl_xor(ls, off, 16);
        a0 += __shfl_xor(a0, off, 16);
        a1 += __shfl_xor(a1, off, 16);
        a2 += __shfl_xor(a2, off, 16);
      }
      if (l16 == g) {                               // one lane per half writes row
        const float inv = 1.0f / ls;
        const int row = i0 + g + 8 * h;
        out[base + 0 * NDIM + row] = x[base + 0 * NDIM + row] + a0 * inv;
        out[base + 1 * NDIM + row] = x[base + 1 * NDIM + row] + a1 * inv;
        out[base + 2 * NDIM + row] = x[base + 2 * NDIM + row] + a2 * inv;
      }
    }
  }
}

// ---------------------------------------------------------------------------
extern "C" void kernel_launch(void* const* d_in, const int* in_sizes, int n_in,
                              void* d_out, int out_size, void* d_ws, size_t ws_size,
                              hipStream_t stream) {
  (void)in_sizes; (void)n_in; (void)out_size; (void)ws_size;
  const float* x  = (const float*)d_in[0];
  const float* Wq = (const float*)d_in[1];
  const float* Wk = (const float*)d_in[2];
  const float* Wv = (const float*)d_in[3];
  float* out = (float*)d_out;
  float* ws  = (float*)d_ws;               // [q | k | v], 3 MB each = 9 MB

  float* qv = ws;
  float* kv = ws + TENSOR_ELEMS;
  float* vv = ws + 2 * (size_t)TENSOR_ELEMS;

  dim3 grid1(MDIM / 64, BDIM, 3);
  vn_qkv_proj_kernel<<<grid1, 256, 0, stream>>>(x, Wq, Wk, Wv, ws);

  dim3 grid2(BDIM * CDIM);
  vn_attn_kernel<<<grid2, 256, 0, stream>>>(x, qv, kv, vv, out);
}